// Resample_82334523064701
// MI455X (gfx1250) — compile-verified
//
#include <hip/hip_runtime.h>

// ---------------------------------------------------------------------------
// torchaudio-style sinc resample for 8 fixed rates, CDNA5 (gfx1250) WMMA f32.
//
// Per batch b: o = orig/gcd, n = new/gcd, width w, taps L = 2w+o,
//   y[c, f*n + j] = sum_l kern[j,l] * x[c, f*o + l - w]   (zero padded x)
//   target = ceil(n*T/o);  downsample batches also write flip: out[2*target-1-m]=y[m]
// Cast as GEMM  Y[F,n] = X[F,L] * K^T  and run on V_WMMA_F32_16X16X4_F32.
// Only the zero-padding tails [len_b, MAXLEN) are memset; the valid region is
// fully covered by the WMMA tiles (minimal store traffic).
// ---------------------------------------------------------------------------

typedef float v2f __attribute__((ext_vector_type(2)));
typedef float v8f __attribute__((ext_vector_type(8)));

#define T_IN   1048576
#define MAXLEN 1992296
#define NB     8
#define LPMAX  36              // max padded tap count (batch 5: L=36)
#define SXMAX  (128 * 20 + 40) // max staged input span: 128*o + Lp, o<=20

struct BP {
    int o, n, w, L, Lp;   // hop, phases, half-width, taps, taps padded to 4
    int target;           // output samples before flip-concat
    int F;                // frames = ceil(target/n)
    int flip;             // 1 => concat(y, reverse(y))
    float base;           // min(o,n) * 0.99
};

__constant__ BP g_bp[NB] = {
    { 5,  4, 8, 21, 24,  838861,  209716, 1,  3.96f},   // k=16
    {20, 23, 7, 34, 36, 1205863,   52429, 0, 19.80f},   // k=23
    { 1,  1, 7, 15, 16, 1048576, 1048576, 0,  0.99f},   // k=20
    {10,  9, 7, 24, 24,  943719,  104858, 1,  8.91f},   // k=18
    {20, 21, 7, 34, 36, 1101005,   52429, 0, 19.80f},   // k=21
    {20, 17, 8, 36, 36,  891290,   52429, 1, 16.83f},   // k=17
    {10, 11, 7, 24, 24, 1153434,  104858, 0,  9.90f},   // k=22
    {20, 19, 7, 34, 36,  996148,   52429, 1, 18.81f},   // k=19
};

// blocks per batch = ceil(F/128): {1639,410,8192,820,410,410,820,410}
__constant__ int g_prefix[NB + 1] = {0, 1639, 2049, 10241, 11061, 11471, 11881, 12701, 13111};
#define TOTAL_BLOCKS 13111

// valid output length per batch: flip ? 2*target : target
__constant__ int g_len[NB] = {
    1677722, 1205863, 1048576, 1887438, 1101005, 1782580, 1153434, 1992296
};
// max padding = MAXLEN - min(len) = 1992296 - 1048576 = 943720 -> 3687 blocks
#define PAD_BLOCKS 3687

__global__ __launch_bounds__(256) void zero_pad(float* __restrict__ out) {
    const int bc = blockIdx.y;             // (batch*2 + channel), 0..15
    const int len = g_len[bc >> 1];
    const int i = len + blockIdx.x * 256 + (int)threadIdx.x;
    if (i < MAXLEN) out[(size_t)bc * MAXLEN + i] = 0.f;
}

__global__ __launch_bounds__(256) void resample_wmma(const float* __restrict__ x,
                                                     float* __restrict__ out) {
    __shared__ float sx[SXMAX];        // staged input window (zero-extended)
    __shared__ float sk[32 * LPMAX];   // filter bank, rows padded to 32, cols to LPMAX

    const int bx = blockIdx.x;
    int b = 0;
#pragma unroll
    for (int i = 1; i < NB; ++i) b += (bx >= g_prefix[i]);
    const BP p = g_bp[b];
    const int c = blockIdx.y;

    // ---- build sinc filter bank in LDS (float32, mirrors the reference) ----
    const float scale = p.base / (float)p.o;
    for (int idx = threadIdx.x; idx < 32 * LPMAX; idx += 256) {
        const int j  = idx / LPMAX;
        const int kk = idx % LPMAX;
        float v = 0.f;
        if (j < p.n && kk < p.L) {
            float fi = (float)(kk - p.w) / (float)p.o;
            float tt = ((float)(-j) / (float)p.n + fi) * p.base;
            tt = fminf(6.f, fmaxf(-6.f, tt));
            float wc = cosf(tt * (float)(M_PI / 12.0));  // hann^2 window
            float wnd = wc * wc;
            float tp = tt * (float)M_PI;
            float s  = (tp == 0.f) ? 1.f : (sinf(tp) / tp);
            v = s * wnd * scale;
        }
        sk[idx] = v;
    }

    // ---- stage input window: x[f*o + kk - w] for f in [fb0, fb0+128) ----
    const int fb0  = (bx - g_prefix[b]) * 128;
    const int g0   = fb0 * p.o - p.w;
    const int span = 128 * p.o + p.Lp;
    const float* __restrict__ xin = x + (size_t)(b * 2 + c) * T_IN;
    for (int i = threadIdx.x; i < span; i += 256) {
        const int g = g0 + i;
        sx[i] = (g >= 0 && g < T_IN) ? xin[g] : 0.f;
    }
    __syncthreads();

    // ---- one 16-frame M-tile per wave, N padded to 16 (or 2x16 if n>16) ----
    const int wv   = threadIdx.x >> 5;
    const int lane = threadIdx.x & 31;
    const int fl0  = wv * 16;        // frame offset within the block tile
    const int f0   = fb0 + fl0;
    if (f0 >= p.F) return;           // wave-uniform; EXEC stays all-ones below

    const int row  = lane & 15;
    const int koff = (lane < 16) ? 0 : 2;      // K sub-offset per lane half
    const int Kt   = p.Lp >> 2;
    const int NT   = (p.n > 16) ? 2 : 1;
    float* __restrict__ outp = out + (size_t)(b * 2 + c) * MAXLEN;

    const int abase = (fl0 + row) * p.o + koff;        // A: lane = M row
    for (int nt = 0; nt < NT; ++nt) {
        const int jcol = nt * 16 + row;                // B: lane = N col
        const float* __restrict__ krow = &sk[jcol * LPMAX + koff];
        v8f acc = {0.f, 0.f, 0.f, 0.f, 0.f, 0.f, 0.f, 0.f};
        for (int t = 0; t < Kt; ++t) {
            v2f A, B;
            A.x = sx[abase + 4 * t];
            A.y = sx[abase + 4 * t + 1];
            B.x = krow[4 * t];
            B.y = krow[4 * t + 1];
            acc = __builtin_amdgcn_wmma_f32_16x16x4_f32(
                false, A, false, B, (short)0, acc, false, false);
        }
        // D layout: VGPR r -> M = r (lanes 0-15) / r+8 (lanes 16-31), N = lane&15
#pragma unroll
        for (int r = 0; r < 8; ++r) {
            const int f = f0 + ((lane < 16) ? r : (r + 8));
            const int m = f * p.n + jcol;
            if (jcol < p.n && m < p.target) {
                outp[m] = acc[r];
                if (p.flip) outp[2 * p.target - 1 - m] = acc[r];
            }
        }
    }
}

extern "C" void kernel_launch(void* const* d_in, const int* in_sizes, int n_in,
                              void* d_out, int out_size, void* d_ws, size_t ws_size,
                              hipStream_t stream) {
    (void)in_sizes; (void)n_in; (void)d_ws; (void)ws_size; (void)out_size;
    const float* x = (const float*)d_in[0];
    float* out = (float*)d_out;

    // Zero only the padding tails; the valid region is fully written below.
    dim3 pgrid(PAD_BLOCKS, 2 * NB, 1);
    zero_pad<<<pgrid, 256, 0, stream>>>(out);

    dim3 grid(TOTAL_BLOCKS, 2, 1);
    resample_wmma<<<grid, 256, 0, stream>>>(x, out);
}